// GCN_27161373180172
// MI455X (gfx1250) — compile-verified
//
#include <hip/hip_runtime.h>

typedef float v2f __attribute__((ext_vector_type(2)));
typedef float v4f __attribute__((ext_vector_type(4)));
typedef float v8f __attribute__((ext_vector_type(8)));

// ---------------- utility kernels ----------------

__global__ void k_fill(float* __restrict__ p, float v, long n) {
    long i = blockIdx.x * (long)blockDim.x + threadIdx.x;
    if (i < n) p[i] = v;
}

__global__ void k_copy4(const v4f* __restrict__ a, v4f* __restrict__ b, long n4) {
    long i = blockIdx.x * (long)blockDim.x + threadIdx.x;
    if (i < n4) b[i] = a[i];
}

// deg[v] = 1 (self loop) + in-degree(v)
__global__ void k_deg_count(const int* __restrict__ dst, float* __restrict__ deg, long ne) {
    long i = blockIdx.x * (long)blockDim.x + threadIdx.x;
    if (i < ne) atomicAdd(&deg[dst[i]], 1.0f);
}

__global__ void k_dinv(const float* __restrict__ deg, float* __restrict__ dinv, long n) {
    long i = blockIdx.x * (long)blockDim.x + threadIdx.x;
    if (i < n) dinv[i] = rsqrtf(deg[i]);
}

// ---------------- WMMA GEMMs (fp32 16x16x4) ----------------
// Layer 1: g[row,0:16] = (x[row,0:128] @ W1[128,16]) * dinv[row]
// A tile (16 rows x 128 cols fp32 = 8KB per wave) is staged into LDS via
// global_load_async_to_lds_b128 (coalesced 512B per issue, ASYNCcnt-tracked),
// then A fragments are read from LDS.
__global__ void __launch_bounds__(256) k_gemm1_wmma(
        const float* __restrict__ x, const float* __restrict__ W,
        const float* __restrict__ dinv, float* __restrict__ g, int nrows) {
    __shared__ float smem[8][16 * 128];      // 8 waves * 8KB = 64KB per block

    int lane = threadIdx.x & 31;
    int wave = threadIdx.x >> 5;
    int tile = blockIdx.x * (blockDim.x >> 5) + wave;
    long base = (long)tile * 16;
    if (base >= nrows) return;               // wave-uniform: EXEC stays all-ones
    int hi = lane >> 4;                      // 0: lanes 0-15, 1: lanes 16-31
    int ln = lane & 15;

    // ---- async-stage this wave's 16x128 tile into LDS ----
    unsigned lds_base = (unsigned)(uintptr_t)&smem[wave][0];
#pragma unroll
    for (int r = 0; r < 16; ++r) {
        long rr = base + r;
        if (rr >= nrows) rr = nrows - 1;     // clamp tail rows (stores guarded later)
        unsigned voff = (unsigned)(rr * 512u) + (unsigned)(lane * 16);  // bytes into x
        unsigned ldsa = lds_base + (unsigned)(r * 512 + lane * 16);
        asm volatile("global_load_async_to_lds_b128 %0, %1, %2 offset:0"
                     :
                     : "v"(ldsa), "v"(voff), "s"(x)
                     : "memory");
    }
    asm volatile("s_wait_asynccnt 0x0" ::: "memory");   // wave-private tile: no barrier

    const float* ar = &smem[wave][ln * 128 + 2 * hi];   // A row ln, K split by lane half

    v8f c = {};
#pragma unroll
    for (int k0 = 0; k0 < 128; k0 += 4) {
        // A 16x4 f32: lane<16 holds (row, k0+0/k0+1); lane>=16 holds (row, k0+2/k0+3)
        v2f a = *(const v2f*)(ar + k0);                 // ds_load_b64
        // B 4x16 f32: VGPR0 = row k0+2*hi, VGPR1 = row k0+2*hi+1; N = ln
        v2f b;
        b.x = W[(k0 + 2 * hi) * 16 + ln];
        b.y = W[(k0 + 2 * hi + 1) * 16 + ln];
        c = __builtin_amdgcn_wmma_f32_16x16x4_f32(false, a, false, b, (short)0, c,
                                                  false, false);
    }
    // D 16x16 f32: VGPR r -> M = r + 8*hi, N = ln
#pragma unroll
    for (int r = 0; r < 8; ++r) {
        long m = base + hi * 8 + r;
        if (m < nrows) g[m * 16 + ln] = c[r] * dinv[m];
    }
}

// Layer 2: g[row,0:16] = (h[row,0:16] @ W2pad[16,16]) * dinv[row]; W2 is 16x10, cols 10..15 = 0
__global__ void k_gemm2_wmma(const float* __restrict__ h, const float* __restrict__ W2,
                             const float* __restrict__ dinv, float* __restrict__ g,
                             int nrows) {
    int lane = threadIdx.x & 31;
    int wave = threadIdx.x >> 5;
    int tile = blockIdx.x * (blockDim.x >> 5) + wave;
    long base = (long)tile * 16;
    if (base >= nrows) return;
    int hi = lane >> 4;
    int ln = lane & 15;
    long row = base + ln;
    if (row >= nrows) row = nrows - 1;
    const float* hr = h + row * 16 + 2 * hi;

    v8f c = {};
#pragma unroll
    for (int k0 = 0; k0 < 16; k0 += 4) {
        v2f a = *(const v2f*)(hr + k0);
        v2f b;
        b.x = (ln < 10) ? W2[(k0 + 2 * hi) * 10 + ln] : 0.0f;
        b.y = (ln < 10) ? W2[(k0 + 2 * hi + 1) * 10 + ln] : 0.0f;
        c = __builtin_amdgcn_wmma_f32_16x16x4_f32(false, a, false, b, (short)0, c,
                                                  false, false);
    }
#pragma unroll
    for (int r = 0; r < 8; ++r) {
        long m = base + hi * 8 + r;
        if (m < nrows) g[m * 16 + ln] = c[r] * dinv[m];
    }
}

// ---------------- edge aggregation ----------------
// 16 consecutive lanes handle the 16 columns of one edge (contiguous 64B per half-wave)
__global__ void k_agg_edges(const float* __restrict__ g, const int* __restrict__ src,
                            const int* __restrict__ dst, float* __restrict__ acc,
                            long ne, int ncols) {
    long t = blockIdx.x * (long)blockDim.x + threadIdx.x;
    long e = t >> 4;
    int c = (int)(t & 15);
    if (e >= ne || c >= ncols) return;
    int s = src[e];
    int d = dst[e];
    atomicAdd(acc + (long)d * 16 + c, g[(long)s * 16 + c]);
}

// h = relu(acc * dinv[row] + b1[col])
__global__ void k_fin1(const float* __restrict__ acc, const float* __restrict__ dinv,
                       const float* __restrict__ b1, float* __restrict__ h, long n16) {
    long i = blockIdx.x * (long)blockDim.x + threadIdx.x;
    if (i >= n16) return;
    long row = i >> 4;
    int c = (int)(i & 15);
    float v = acc[i] * dinv[row] + b1[c];
    h[i] = v > 0.0f ? v : 0.0f;
}

// finalize layer 2 + global mean-pool accumulation
__global__ void k_fin2_pool(const float* __restrict__ acc, const float* __restrict__ dinv,
                            const float* __restrict__ b2, const int* __restrict__ batch,
                            float* __restrict__ pooled, float* __restrict__ cnt,
                            long nnodes) {
    long t = blockIdx.x * (long)blockDim.x + threadIdx.x;
    long node = t >> 4;
    int c = (int)(t & 15);
    if (node >= nnodes) return;
    int grp = batch[node];
    if (c < 10) {
        float v = acc[node * 16 + c] * dinv[node] + b2[c];
        atomicAdd(pooled + (long)grp * 10 + c, v);
    } else if (c == 15) {
        atomicAdd(cnt + grp, 1.0f);
    }
}

// per-graph mean + log_softmax over 10 classes
__global__ void k_logsoftmax(const float* __restrict__ pooled, const float* __restrict__ cnt,
                             float* __restrict__ out, int ngraphs) {
    int gi = blockIdx.x * blockDim.x + threadIdx.x;
    if (gi >= ngraphs) return;
    float n = cnt[gi];
    n = n > 1.0f ? n : 1.0f;
    float m[10];
    float mx = -3.0e38f;
#pragma unroll
    for (int c = 0; c < 10; ++c) {
        m[c] = pooled[gi * 10 + c] / n;
        mx = fmaxf(mx, m[c]);
    }
    float s = 0.0f;
#pragma unroll
    for (int c = 0; c < 10; ++c) s += expf(m[c] - mx);
    float l = logf(s);
#pragma unroll
    for (int c = 0; c < 10; ++c) out[gi * 10 + c] = m[c] - mx - l;
}

// ---------------- launch ----------------

extern "C" void kernel_launch(void* const* d_in, const int* in_sizes, int n_in,
                              void* d_out, int out_size, void* d_ws, size_t ws_size,
                              hipStream_t stream) {
    const float* x    = (const float*)d_in[0];
    const float* W1   = (const float*)d_in[1];
    const float* b1   = (const float*)d_in[2];
    const float* W2   = (const float*)d_in[3];
    const float* b2   = (const float*)d_in[4];
    const int*   eidx = (const int*)d_in[5];
    const int*   batch= (const int*)d_in[6];

    long N = in_sizes[0] / 128;      // nodes
    long E = in_sizes[5] / 2;        // edges
    const int* src = eidx;           // edge_index[0]
    const int* dst = eidx + E;       // edge_index[1]
    int G = out_size / 10;           // graphs

    // workspace layout (floats)
    float* ws = (float*)d_ws;
    float* degf   = ws;  ws += N;
    float* dinv   = ws;  ws += N;
    float* g1     = ws;  ws += N * 16;   // reused as g2
    float* acc1   = ws;  ws += N * 16;   // reused as acc2
    float* h1     = ws;  ws += N * 16;
    float* pooled = ws;  ws += (long)G * 10;
    float* cnt    = ws;  ws += G;        // contiguous after pooled

    const int B = 256;
    long n16 = N * 16;

    // degrees + dinv
    k_fill<<<(int)((N + B - 1) / B), B, 0, stream>>>(degf, 1.0f, N);
    k_fill<<<(int)(((long)G * 11 + B - 1) / B), B, 0, stream>>>(pooled, 0.0f, (long)G * 11);
    k_deg_count<<<(int)((E + B - 1) / B), B, 0, stream>>>(dst, degf, E);
    k_dinv<<<(int)((N + B - 1) / B), B, 0, stream>>>(degf, dinv, N);

    int tiles = (int)((N + 15) / 16);
    int gemm_blocks = (tiles + 7) / 8;   // 8 waves (128 rows) per 256-thread block

    // layer 1
    k_gemm1_wmma<<<gemm_blocks, 256, 0, stream>>>(x, W1, dinv, g1, (int)N);
    k_copy4<<<(int)((N * 4 + B - 1) / B), B, 0, stream>>>((const v4f*)g1, (v4f*)acc1, N * 4);
    k_agg_edges<<<(int)((E * 16 + B - 1) / B), B, 0, stream>>>(g1, src, dst, acc1, E, 16);
    k_fin1<<<(int)((n16 + B - 1) / B), B, 0, stream>>>(acc1, dinv, b1, h1, n16);

    // layer 2 (reuse g1/acc1 buffers)
    k_gemm2_wmma<<<gemm_blocks, 256, 0, stream>>>(h1, W2, dinv, g1, (int)N);
    k_copy4<<<(int)((N * 4 + B - 1) / B), B, 0, stream>>>((const v4f*)g1, (v4f*)acc1, N * 4);
    k_agg_edges<<<(int)((E * 16 + B - 1) / B), B, 0, stream>>>(g1, src, dst, acc1, E, 10);

    // pool + log_softmax
    k_fin2_pool<<<(int)((n16 + B - 1) / B), B, 0, stream>>>(acc1, dinv, b2, batch,
                                                            pooled, cnt, N);
    k_logsoftmax<<<(G + B - 1) / B, B, 0, stream>>>(pooled, cnt, (float*)d_out, G);
}